// RangeToBEV_24498493456757
// MI455X (gfx1250) — compile-verified
//
#include <hip/hip_runtime.h>
#include <hip/hip_bf16.h>

// ---------------------------------------------------------------- constants
#define B_    2
#define C_    64
#define H_    64
#define W_    128
#define HW_   (H_ * W_)        // 8192
#define NX_   512
#define NY_   512
#define NYNX_ (NY_ * NX_)      // 262144
#define BIGD  1e10f
#define INV_VOX 10.0f          // 1/0.1
#define X0_   0.0f
#define Y0_   (-25.6f)

typedef float v2f __attribute__((ext_vector_type(2)));
typedef float v8f __attribute__((ext_vector_type(8)));

// ---------------------------------------------------------------- zero
__global__ void rb_zero(float* __restrict__ p, size_t n) {
    size_t i = (size_t)blockIdx.x * blockDim.x + threadIdx.x;
    size_t stride = (size_t)gridDim.x * blockDim.x;
    for (; i < n; i += stride) p[i] = 0.0f;
}

// ---------------------------------------------------------------- pack near points:
// near4[b][n] = (x,y,z,|n|^2) if valid else (0,0,0,BIG)  -> d2 row == BIG (ref mask)
__global__ void rb_prep_near(const float* __restrict__ pts,  // (B,4,H,W)
                             const int* __restrict__ mask,   // (B,H,W)
                             float4* __restrict__ near4) {
    int t = blockIdx.x * blockDim.x + threadIdx.x;           // B*HW
    if (t >= B_ * HW_) return;
    int b = t / HW_, n = t % HW_;
    const float* p = pts + (size_t)b * 4 * HW_;
    float x = p[n], y = p[HW_ + n], z = p[2 * HW_ + n];
    float4 o;
    if (mask[t] > 0) o = make_float4(x, y, z, x * x + y * y + z * z);
    else             o = make_float4(0.f, 0.f, 0.f, BIGD);
    near4[t] = o;
}

// ---------------------------------------------------------------- WMMA 3-NN
// Wave handles 16 far points (N axis). Per WMMA step: 16 near points (M axis).
// A (16x4 f32): lanes 0-15 give (K0,K1)=(x,y) of near point laneN,
//               lanes 16-31 give (K2,K3)=(z, |n|^2|BIG) of the same point.
// B (4x16 f32): lanes 0-15 give (K0,K1)=(-2fx,-2fy) of far point laneN,
//               lanes 16-31 give (K2,K3)=(-2fz, 1).
// D[m][n] = |n|^2 - 2 f.n ; per-far |f|^2 added after top-3 (order-invariant).
#define TILE_ 1024

// Branchless sorted-3 insert: 3 v_cmp + ~10 v_cndmask, no exec divergence.
#define INSERT3(val, nidx)                                                   \
    do {                                                                     \
        const float _v = (val); const int _i = (nidx);                       \
        const bool c0 = _v < d0v, c1 = _v < d1v, c2 = _v < d2v;              \
        const float n2 = c1 ? d1v : (c2 ? _v : d2v);                         \
        const int   m2 = c1 ? i1  : (c2 ? _i : i2);                          \
        const float n1 = c0 ? d0v : (c1 ? _v : d1v);                         \
        const int   m1 = c0 ? i0  : (c1 ? _i : i1);                          \
        const float n0 = c0 ? _v : d0v;                                      \
        const int   m0 = c0 ? _i : i0;                                       \
        d2v = n2; i2 = m2; d1v = n1; i1 = m1; d0v = n0; i0 = m0;             \
    } while (0)

// min3-reduce 8 WMMA outputs, wave-uniform skip, branchless insert of survivors.
#define PROC8(ACC, BASE)                                                     \
    do {                                                                     \
        const float mn =                                                     \
            fminf(fminf(fminf((ACC)[0], (ACC)[1]), fminf((ACC)[2], (ACC)[3])),\
                  fminf(fminf((ACC)[4], (ACC)[5]), fminf((ACC)[6], (ACC)[7])));\
        if (__any(mn < d2v)) {                                               \
            const int _b8 = (BASE);                                          \
            _Pragma("unroll")                                                \
            for (int r = 0; r < 8; ++r) INSERT3((ACC)[r], _b8 + r);          \
        }                                                                    \
    } while (0)

__launch_bounds__(256)
__global__ void rb_topk_wmma(const float4* __restrict__ near4g, // (B,HW)
                             const float*  __restrict__ far_pts,// (B,4,H,W)
                             float4* __restrict__ w4out,        // (B,HW)
                             int4*   __restrict__ idx4out) {    // (B,HW)
    __shared__ float4 sNear[TILE_];
    const int b     = blockIdx.y;
    const int lane  = threadIdx.x & 31;
    const int wave  = threadIdx.x >> 5;
    const int laneN = lane & 15;
    const int half  = lane >> 4;

    const int far = blockIdx.x * 128 + wave * 16 + laneN;      // this lane's far point
    const float* pf = far_pts + (size_t)b * 4 * HW_;
    const float fx = pf[far], fy = pf[HW_ + far], fz = pf[2 * HW_ + far];

    v2f bv;
    if (half == 0) { bv[0] = -2.f * fx; bv[1] = -2.f * fy; }
    else           { bv[0] = -2.f * fz; bv[1] = 1.f; }
    const float f2 = fx * fx + fy * fy + fz * fz;

    float d0v = 3.4e38f, d1v = 3.4e38f, d2v = 3.4e38f;
    int   i0 = 0, i1 = 0, i2 = 0;

    const float4* ng = near4g + (size_t)b * HW_;
    const v2f* s2 = (const v2f*)sNear;               // lane's A pair as one ds_load_b64

    for (int t0 = 0; t0 < HW_; t0 += TILE_) {
        for (int i = threadIdx.x; i < TILE_; i += 256) sNear[i] = ng[t0 + i];
        if (t0 + TILE_ < HW_)
            __builtin_prefetch(&ng[t0 + TILE_ + threadIdx.x], 0, 1); // global_prefetch
        __syncthreads();
        // 2x unrolled + double-issued LDS loads: first WMMA waits dscnt<=1,
        // second load's latency hides under the first WMMA + select block.
        for (int j = 0; j < TILE_; j += 32) {
            const v2f av0 = s2[((j + laneN) << 1) | half];
            const v2f av1 = s2[((j + 16 + laneN) << 1) | half];
            v8f c0 = {}, c1 = {};
            v8f acc0 = __builtin_amdgcn_wmma_f32_16x16x4_f32(
                false, av0, false, bv, (short)0, c0, false, false);
            v8f acc1 = __builtin_amdgcn_wmma_f32_16x16x4_f32(
                false, av1, false, bv, (short)0, c1, false, false);
            PROC8(acc0, t0 + j + half * 8);
            PROC8(acc1, t0 + j + 16 + half * 8);
        }
        __syncthreads();
    }

    // merge the two M-halves (lane L <-> lane L+16 hold the same far point)
    float e0 = __shfl_xor(d0v, 16, 32), e1 = __shfl_xor(d1v, 16, 32), e2 = __shfl_xor(d2v, 16, 32);
    int   j0 = __shfl_xor(i0, 16, 32),  j1 = __shfl_xor(i1, 16, 32),  j2 = __shfl_xor(i2, 16, 32);
    INSERT3(e0, j0); INSERT3(e1, j1); INSERT3(e2, j2);

    if (half == 0) {
        const float q0 = d0v + f2, q1 = d1v + f2, q2 = d2v + f2; // true squared dists
        const float r0 = 1.f / (q0 + 1e-8f), r1 = 1.f / (q1 + 1e-8f), r2 = 1.f / (q2 + 1e-8f);
        const float s = r0 + r1 + r2;
        w4out[(size_t)b * HW_ + far]   = make_float4(r0 / s, r1 / s, r2 / s, 0.f);
        idx4out[(size_t)b * HW_ + far] = make_int4(i0, i1, i2, 0);
    }
}

// ---------------------------------------------------------------- interpolation gather
__global__ void rb_interp(const float* __restrict__ fv,       // (B,C,H,W)
                          const float4* __restrict__ w4,
                          const int4*   __restrict__ idx4,
                          float* __restrict__ interp) {       // (B,HW,C)
    int t = blockIdx.x * blockDim.x + threadIdx.x;            // B*HW*C
    if (t >= B_ * HW_ * C_) return;
    const int c  = t & (C_ - 1);
    const int pm = t >> 6;          // b*HW + m
    const int b  = pm / HW_;
    const float4 w = w4[pm];
    const int4  ii = idx4[pm];
    const float* f = fv + ((size_t)b * C_ + c) * HW_;
    interp[t] = w.x * f[ii.x] + w.y * f[ii.y] + w.z * f[ii.z];
}

// ---------------------------------------------------------------- mean-scatter into BEV
__global__ void rb_scatter(const float* __restrict__ fv,
                           const float* __restrict__ near_pts, const int* __restrict__ mask,
                           const float* __restrict__ far_pts,  const int* __restrict__ mask_far,
                           const float* __restrict__ interp,
                           float* __restrict__ out,            // (B,C,NY,NX) sums
                           float* __restrict__ cnt) {          // (B,NY*NX)
    int t = blockIdx.x * blockDim.x + threadIdx.x;             // B*2*HW
    if (t >= B_ * 2 * HW_) return;
    const int b = t / (2 * HW_);
    const int p = t % (2 * HW_);
    float x, y; int valid;
    if (p < HW_) {
        const float* pp = near_pts + (size_t)b * 4 * HW_;
        x = pp[p]; y = pp[HW_ + p]; valid = mask[b * HW_ + p];
    } else {
        const int m = p - HW_;
        const float* pp = far_pts + (size_t)b * 4 * HW_;
        x = pp[m]; y = pp[HW_ + m]; valid = mask_far[b * HW_ + m];
    }
    const int ix = (int)floorf((x - X0_) * INV_VOX);
    const int iy = (int)floorf((y - Y0_) * INV_VOX);
    if (valid > 0 && ix >= 0 && ix < NX_ && iy >= 0 && iy < NY_) {
        const int flat = iy * NX_ + ix;
        atomicAdd(cnt + (size_t)b * NYNX_ + flat, 1.0f);
        float* ob = out + (size_t)b * C_ * NYNX_ + flat;
        if (p < HW_) {
            const float* f = fv + (size_t)b * C_ * HW_ + p;
            for (int c = 0; c < C_; ++c) atomicAdd(ob + (size_t)c * NYNX_, f[(size_t)c * HW_]);
        } else {
            const float* f = interp + ((size_t)b * HW_ + (p - HW_)) * C_;
            for (int c = 0; c < C_; ++c) atomicAdd(ob + (size_t)c * NYNX_, f[c]);
        }
    }
}

// ---------------------------------------------------------------- divide by count
__global__ void rb_finalize(float* __restrict__ out, const float* __restrict__ cnt) {
    size_t t = (size_t)blockIdx.x * blockDim.x + threadIdx.x;  // B*C*NYNX
    if (t >= (size_t)B_ * C_ * NYNX_) return;
    const int p  = (int)(t % NYNX_);
    const int bc = (int)(t / NYNX_);
    const int b  = bc / C_;
    out[t] = out[t] / fmaxf(cnt[(size_t)b * NYNX_ + p], 1.0f);
}

// ---------------------------------------------------------------- launch
extern "C" void kernel_launch(void* const* d_in, const int* in_sizes, int n_in,
                              void* d_out, int out_size, void* d_ws, size_t ws_size,
                              hipStream_t stream) {
    const float* fv   = (const float*)d_in[0];   // fv_features (B,C,H,W)
    const float* pim  = (const float*)d_in[1];   // points_img  (B,4,H,W)
    const int*   msk  = (const int*)  d_in[2];   // proj_masks  (B,H,W)
    const float* pimf = (const float*)d_in[3];   // points_img_far
    const int*   mskf = (const int*)  d_in[4];   // proj_masks_far
    float* out = (float*)d_out;                  // (B,C,NY,NX)

    float*  ws     = (float*)d_ws;
    float4* near4  = (float4*)ws;                                // B*HW float4
    float4* w4     = (float4*)(ws + (size_t)B_ * HW_ * 4);       // B*HW float4
    int4*   idx4   = (int4*)  (ws + (size_t)2 * B_ * HW_ * 4);   // B*HW int4
    float*  interp =           ws + (size_t)3 * B_ * HW_ * 4;    // B*HW*C
    float*  cnt    =  interp + (size_t)B_ * HW_ * C_;            // B*NYNX

    const size_t nOut = (size_t)B_ * C_ * NYNX_;
    rb_zero<<<(int)((nOut + 255) / 256), 256, 0, stream>>>(out, nOut);
    rb_zero<<<(int)(((size_t)B_ * NYNX_ + 255) / 256), 256, 0, stream>>>(cnt, (size_t)B_ * NYNX_);

    rb_prep_near<<<(B_ * HW_ + 255) / 256, 256, 0, stream>>>(pim, msk, near4);

    dim3 gTop(HW_ / 128, B_);
    rb_topk_wmma<<<gTop, 256, 0, stream>>>(near4, pimf, w4, idx4);

    rb_interp<<<(B_ * HW_ * C_ + 255) / 256, 256, 0, stream>>>(fv, w4, idx4, interp);

    rb_scatter<<<(B_ * 2 * HW_ + 255) / 256, 256, 0, stream>>>(
        fv, pim, msk, pimf, mskf, interp, out, cnt);

    rb_finalize<<<(int)((nOut + 255) / 256), 256, 0, stream>>>(out, cnt);
}